// HierarchyAttention_28269474742692
// MI455X (gfx1250) — compile-verified
//
#include <hip/hip_runtime.h>
#include <math.h>

// ---------------------------------------------------------------------------
// CDNA5 (gfx1250) WMMA f32 16x16x4 helpers.
//   A: 16x4 f32  -> 2 VGPRs/lane: lane<16 holds A[m][0],A[m][1]; lane>=16 A[m][2],A[m][3]
//   B: 4x16 f32  -> 2 VGPRs/lane: lane<16 holds B[0][n],B[1][n]; lane>=16 B[2][n],B[3][n]
//   C/D: 16x16 f32 -> 8 VGPRs/lane: elem rr -> M = rr + 8*(lane>>4), N = lane&15
// ---------------------------------------------------------------------------
typedef __attribute__((ext_vector_type(2))) float v2f;
typedef __attribute__((ext_vector_type(8))) float v8f;

__device__ __forceinline__ v8f wmma_f32_k4(v8f acc, float a0, float a1, float b0, float b1) {
  v2f A; A[0] = a0; A[1] = a1;
  v2f B; B[0] = b0; B[1] = b1;
  // 8 args: (neg_a, A, neg_b, B, c_mod, C, reuse_a, reuse_b)
  return __builtin_amdgcn_wmma_f32_16x16x4_f32(false, A, false, B, (short)0, acc, false, false);
}

__device__ __forceinline__ float sigmoidf_(float x) { return 1.0f / (1.0f + __expf(-x)); }

// ---------------------------------------------------------------------------
// Input-gate GEMM: out[n, g] = sum_k X[n,k] * W[g,k] + bias[g], g in [0,384)
// Row n = seq*T + t. X row is either emb[tokens[seq*T + src_t]] (gather) or
// Xdense[seq*T + src_t]. src_t = t (fwd) or clip(len-1-t, 0) (bwd reversal).
// Block: 256 thr (8 waves). Block tile: 16 rows x 128 cols (blockIdx.y: 3 col groups).
// ---------------------------------------------------------------------------
__global__ void xg_gemm_kernel(const float* __restrict__ W,      // [384, K]
                               const float* __restrict__ bias,   // [384]
                               const float* __restrict__ emb,    // [V, K] or null
                               const int*   __restrict__ tokens, // [nseq*T] or null
                               const float* __restrict__ Xdense, // [nseq*T, K] or null
                               const int*   __restrict__ lens,   // [nseq]
                               float* __restrict__ out,          // [nseq*T, 384]
                               int T, int K, int reverse)
{
  __shared__ float As[16 * 256];
  const int tid = threadIdx.x;
  const int rowBase = blockIdx.x * 16;
  const int colBase = blockIdx.y * 128;

  // Stage 16 x K A-panel (with embedding gather and optional reversal)
  for (int i = tid; i < 16 * K; i += 256) {
    int r = i / K, k = i - r * K;
    int row = rowBase + r;
    int seq = row / T, t = row - seq * T;
    int st = t;
    if (reverse) {
      int len = lens[seq]; if (len < 1) len = 1;
      st = len - 1 - t; if (st < 0) st = 0;
    }
    float v;
    if (tokens) {
      int tok = tokens[seq * T + st];
      v = emb[(size_t)tok * K + k];
    } else {
      v = Xdense[(size_t)(seq * T + st) * K + k];
    }
    As[r * K + k] = v;
  }
  __syncthreads();

  const int wave = tid >> 5;
  const int lane = tid & 31;
  const int m    = lane & 15;   // A row / B col index for this lane
  const int half = lane >> 4;
  const int gcol = colBase + wave * 16 + m;  // output column / weight row

  v8f acc = {};
  for (int kk = 0; kk < K; kk += 4) {
    int k0 = kk + 2 * half;
    float2 a = *reinterpret_cast<const float2*>(&As[m * K + k0]);
    float2 b = *reinterpret_cast<const float2*>(&W[(size_t)gcol * K + k0]);
    acc = wmma_f32_k4(acc, a.x, a.y, b.x, b.y);
  }
  const float bv = bias[gcol];
#pragma unroll
  for (int rr = 0; rr < 8; ++rr) {
    int mr = rr + 8 * half;
    out[(size_t)(rowBase + mr) * 384 + gcol] = acc[rr] + bv;
  }
}

// ---------------------------------------------------------------------------
// GRU sequential scan over T steps for 16 sequences per block.
// h state [16,128] lives in LDS. Per step: gh = h @ whh^T (+bhh) via WMMA
// (wave w owns hidden cols [w*16, w*16+16) for all three gates), then the
// elementwise GRU update. Backward direction scatters h to reversed slots.
// ---------------------------------------------------------------------------
__global__ void gru_scan_kernel(const float* __restrict__ xg,   // [nseq*T, 384]
                                const float* __restrict__ whh,  // [384, 128]
                                const float* __restrict__ bhh,  // [384]
                                const int*   __restrict__ lens, // [nseq]
                                float* __restrict__ hout,       // [nseq, T, 256]
                                int T, int colOffset, int reverse)
{
  __shared__ float hbuf[16 * 128];
  __shared__ int lenbuf[16];
  const int tid = threadIdx.x;
  const int seqBase = blockIdx.x * 16;

  for (int i = tid; i < 16 * 128; i += 256) hbuf[i] = 0.0f;
  if (tid < 16) { int l = lens[seqBase + tid]; lenbuf[tid] = (l < 1) ? 1 : l; }
  __syncthreads();

  const int wave = tid >> 5;
  const int lane = tid & 31;
  const int n    = lane & 15;
  const int half = lane >> 4;
  const int col  = wave * 16 + n;  // hidden column this lane produces
  const float bR = bhh[col], bZ = bhh[128 + col], bN = bhh[256 + col];

  for (int t = 0; t < T; ++t) {
    v8f aR = {}, aZ = {}, aN = {};
    for (int kk = 0; kk < 128; kk += 4) {
      int k0 = kk + 2 * half;
      float2 a  = *reinterpret_cast<const float2*>(&hbuf[n * 128 + k0]);
      float2 br = *reinterpret_cast<const float2*>(&whh[(size_t)(col) * 128 + k0]);
      float2 bz = *reinterpret_cast<const float2*>(&whh[(size_t)(128 + col) * 128 + k0]);
      float2 bn = *reinterpret_cast<const float2*>(&whh[(size_t)(256 + col) * 128 + k0]);
      aR = wmma_f32_k4(aR, a.x, a.y, br.x, br.y);
      aZ = wmma_f32_k4(aZ, a.x, a.y, bz.x, bz.y);
      aN = wmma_f32_k4(aN, a.x, a.y, bn.x, bn.y);
    }
    __syncthreads();  // all reads of hbuf for this step complete
#pragma unroll
    for (int rr = 0; rr < 8; ++rr) {
      int m = rr + 8 * half;   // sequence within tile
      int seq = seqBase + m;
      size_t rowb = ((size_t)seq * T + t) * 384;
      float gir = xg[rowb + col];
      float giz = xg[rowb + 128 + col];
      float gin = xg[rowb + 256 + col];
      float r  = sigmoidf_(gir + aR[rr] + bR);
      float z  = sigmoidf_(giz + aZ[rr] + bZ);
      float nn = tanhf(gin + r * (aN[rr] + bN));
      float hold = hbuf[m * 128 + col];
      float hnew = (1.0f - z) * nn + z * hold;
      hbuf[m * 128 + col] = hnew;   // each (m,col) owned by exactly one lane-elem
      int len = lenbuf[m];
      int dst = reverse ? ((t < len) ? (len - 1 - t) : t) : t;
      hout[((size_t)seq * T + dst) * 256 + colOffset + col] = hnew;
      if (t + 1 < T) __builtin_prefetch(&xg[rowb + 384], 0, 0);  // global_prefetch_b8
    }
    __syncthreads();  // h writes visible before next step's WMMA reads
  }
}

// ---------------------------------------------------------------------------
// Attention pool per sequence: proj = relu(h @ pw^T + pb); score = proj @ ctx;
// masked softmax over t < len; out = sum_t a[t] * h[t,:].
// One block per sequence; 8 waves each own 16 proj columns (C=128).
// ---------------------------------------------------------------------------
__global__ void attn_pool_kernel(const float* __restrict__ h,   // [nseq, T, 256]
                                 const int*   __restrict__ lens,
                                 const float* __restrict__ pw,  // [128, 256]
                                 const float* __restrict__ pb,  // [128]
                                 const float* __restrict__ ctx, // [128]
                                 float* __restrict__ out,       // [nseq, 256]
                                 int T)
{
  __shared__ float Ap[16 * 256];
  __shared__ float sc[64];
  __shared__ float sinv;
  const int tid = threadIdx.x;
  const int seq = blockIdx.x;
  if (tid < T) sc[tid] = 0.0f;
  __syncthreads();

  const int wave = tid >> 5, lane = tid & 31;
  const int m = lane & 15, half = lane >> 4;
  const int col = wave * 16 + m;
  const float bcol = pb[col], ccol = ctx[col];

  for (int rt = 0; rt < T / 16; ++rt) {
    for (int i = tid; i < 16 * 256; i += 256) {
      int r = i >> 8, k = i & 255;
      Ap[i] = h[((size_t)seq * T + rt * 16 + r) * 256 + k];
    }
    __syncthreads();
    v8f acc = {};
    for (int kk = 0; kk < 256; kk += 4) {
      int k0 = kk + 2 * half;
      float2 a = *reinterpret_cast<const float2*>(&Ap[m * 256 + k0]);
      float2 b = *reinterpret_cast<const float2*>(&pw[(size_t)col * 256 + k0]);
      acc = wmma_f32_k4(acc, a.x, a.y, b.x, b.y);
    }
#pragma unroll
    for (int rr = 0; rr < 8; ++rr) {
      float v = acc[rr] + bcol;
      v = v > 0.0f ? v : 0.0f;
      int t = rt * 16 + rr + 8 * half;
      atomicAdd(&sc[t], v * ccol);   // ds_add_f32
    }
    __syncthreads();
  }

  if (tid == 0) {
    int len = lens[seq]; if (len < 1) len = 1; if (len > T) len = T;
    float mx = -3.0e38f;
    for (int t = 0; t < len; ++t) mx = fmaxf(mx, sc[t]);
    float s = 0.0f;
    for (int t = 0; t < T; ++t) {
      float e = (t < len) ? __expf(sc[t] - mx) : 0.0f;
      sc[t] = e; s += e;
    }
    sinv = 1.0f / s;
  }
  __syncthreads();

  float accd = 0.0f;
  for (int t = 0; t < T; ++t) accd += sc[t] * h[((size_t)seq * T + t) * 256 + tid];
  out[(size_t)seq * 256 + tid] = accd * sinv;
}

// ---------------------------------------------------------------------------
// hid = relu(concat(msg_vec, sent_vec) @ fc_w^T + fc_b)  [64,128], K = 512
// ---------------------------------------------------------------------------
__global__ void fc_hid_kernel(const float* __restrict__ msgv,  // [64,256]
                              const float* __restrict__ sentv, // [64,256]
                              const float* __restrict__ fcw,   // [128,512]
                              const float* __restrict__ fcb,   // [128]
                              float* __restrict__ hid)         // [64,128]
{
  __shared__ float Ap[16 * 512];
  const int tid = threadIdx.x;
  const int wave = tid >> 5, lane = tid & 31;
  const int m = lane & 15, half = lane >> 4;
  const int col = wave * 16 + m;
  const float bv = fcb[col];
  for (int rt = 0; rt < 4; ++rt) {
    int rowBase = rt * 16;
    for (int i = tid; i < 16 * 512; i += 256) {
      int r = i >> 9, k = i & 511;
      Ap[i] = (k < 256) ? msgv[(rowBase + r) * 256 + k]
                        : sentv[(rowBase + r) * 256 + (k - 256)];
    }
    __syncthreads();
    v8f acc = {};
    for (int kk = 0; kk < 512; kk += 4) {
      int k0 = kk + 2 * half;
      float2 a = *reinterpret_cast<const float2*>(&Ap[m * 512 + k0]);
      float2 b = *reinterpret_cast<const float2*>(&fcw[(size_t)col * 512 + k0]);
      acc = wmma_f32_k4(acc, a.x, a.y, b.x, b.y);
    }
#pragma unroll
    for (int rr = 0; rr < 8; ++rr) {
      int mr = rowBase + rr + 8 * half;
      float v = acc[rr] + bv;
      hid[mr * 128 + col] = v > 0.0f ? v : 0.0f;
    }
    __syncthreads();
  }
}

__global__ void logits_kernel(const float* __restrict__ hid,   // [64,128]
                              const float* __restrict__ likw,  // [5,128]
                              const float* __restrict__ likb,  // [5]
                              float* __restrict__ out)         // [64,5]
{
  int i = threadIdx.x;  // 320 threads
  int b = i / 5, c = i - b * 5;
  float acc = likb[c];
  for (int j = 0; j < 128; ++j) acc += hid[b * 128 + j] * likw[c * 128 + j];
  out[b * 5 + c] = acc;
}

// ---------------------------------------------------------------------------
// Host launcher
// ---------------------------------------------------------------------------
extern "C" void kernel_launch(void* const* d_in, const int* in_sizes, int n_in,
                              void* d_out, int out_size, void* d_ws, size_t ws_size,
                              hipStream_t stream)
{
  (void)out_size;
  if (n_in < 32) return;

  auto F = [&](int i) { return (const float*)d_in[i]; };
  auto I = [&](int i) { return (const int*)d_in[i]; };

  const float *emb, *fc_b, *fc_w, *lik_b, *lik_w, *msg_ctx, *msg_proj_b, *msg_proj_w;
  const float *s2mb_bhh, *s2mb_bih, *s2mb_whh, *s2mb_wih;
  const float *s2mf_bhh, *s2mf_bih, *s2mf_whh, *s2mf_wih;
  const float *sen_ctx, *sen_proj_b, *sen_proj_w;
  const float *w2sb_bhh, *w2sb_bih, *w2sb_whh, *w2sb_wih;
  const float *w2sf_bhh, *w2sf_bih, *w2sf_whh, *w2sf_wih;
  const int *doc_lens, *doc_sen_lens, *doc_tokens, *sent_lens, *sent_tokens;

  // Layout detection: JAX tree_flatten sorts dict keys alphabetically; a naive
  // harness preserves insertion order. Disambiguate with size sentinels.
  bool alpha = (in_sizes[0] == 64) && (in_sizes[3] == 10000000);
  if (alpha) {
    doc_lens = I(0); doc_sen_lens = I(1); doc_tokens = I(2);
    emb = F(3); fc_b = F(4); fc_w = F(5); lik_b = F(6); lik_w = F(7);
    msg_ctx = F(8); msg_proj_b = F(9); msg_proj_w = F(10);
    s2mb_bhh = F(11); s2mb_bih = F(12); s2mb_whh = F(13); s2mb_wih = F(14);
    s2mf_bhh = F(15); s2mf_bih = F(16); s2mf_whh = F(17); s2mf_wih = F(18);
    sen_ctx = F(19); sen_proj_b = F(20); sen_proj_w = F(21);
    w2sb_bhh = F(22); w2sb_bih = F(23); w2sb_whh = F(24); w2sb_wih = F(25);
    w2sf_bhh = F(26); w2sf_bih = F(27); w2sf_whh = F(28); w2sf_wih = F(29);
    sent_lens = I(30); sent_tokens = I(31);
  } else {
    emb = F(0);
    w2sf_wih = F(1); w2sf_whh = F(2); w2sf_bih = F(3); w2sf_bhh = F(4);
    w2sb_wih = F(5); w2sb_whh = F(6); w2sb_bih = F(7); w2sb_bhh = F(8);
    sen_proj_w = F(9); sen_proj_b = F(10); sen_ctx = F(11);
    s2mf_wih = F(12); s2mf_whh = F(13); s2mf_bih = F(14); s2mf_bhh = F(15);
    s2mb_wih = F(16); s2mb_whh = F(17); s2mb_bih = F(18); s2mb_bhh = F(19);
    msg_proj_w = F(20); msg_proj_b = F(21); msg_ctx = F(22);
    fc_w = F(23); fc_b = F(24); lik_w = F(25); lik_b = F(26);
    sent_tokens = I(27); sent_lens = I(28);
    doc_tokens = I(29); doc_sen_lens = I(30); doc_lens = I(31);
  }

  // Workspace layout (floats)
  float* ws = (float*)d_ws;
  float* xg_buf   = ws;                       // 49152*384 = 18,874,368 (reused)
  float* h_doc    = xg_buf + 18874368;        // 49152*256 = 12,582,912
  float* h_sent   = h_doc + 12582912;         // 4096*256  =  1,048,576
  float* h_msg    = h_sent + 1048576;         // 1024*256  =    262,144
  float* d_vec    = h_msg + 262144;           // 1024*256  =    262,144
  float* sent_vec = d_vec + 262144;           // 64*256    =     16,384
  float* msg_vec  = sent_vec + 16384;         // 64*256    =     16,384
  float* hid      = msg_vec + 16384;          // 64*128    =      8,192
  if ((size_t)33081088 * sizeof(float) > ws_size) return;

  dim3 blk(256);

  // ---- word-level BiGRU on the standalone sentence batch (64 seqs, T=64, K=200)
  xg_gemm_kernel<<<dim3(256, 3), blk, 0, stream>>>(w2sf_wih, w2sf_bih, emb, sent_tokens,
                                                   nullptr, sent_lens, xg_buf, 64, 200, 0);
  gru_scan_kernel<<<dim3(4), blk, 0, stream>>>(xg_buf, w2sf_whh, w2sf_bhh, sent_lens,
                                               h_sent, 64, 0, 0);
  xg_gemm_kernel<<<dim3(256, 3), blk, 0, stream>>>(w2sb_wih, w2sb_bih, emb, sent_tokens,
                                                   nullptr, sent_lens, xg_buf, 64, 200, 1);
  gru_scan_kernel<<<dim3(4), blk, 0, stream>>>(xg_buf, w2sb_whh, w2sb_bhh, sent_lens,
                                               h_sent, 64, 128, 1);
  attn_pool_kernel<<<dim3(64), blk, 0, stream>>>(h_sent, sent_lens, sen_proj_w, sen_proj_b,
                                                 sen_ctx, sent_vec, 64);

  // ---- word-level BiGRU on all doc sentences (1024 seqs, T=48, K=200)
  xg_gemm_kernel<<<dim3(3072, 3), blk, 0, stream>>>(w2sf_wih, w2sf_bih, emb, doc_tokens,
                                                    nullptr, doc_sen_lens, xg_buf, 48, 200, 0);
  gru_scan_kernel<<<dim3(64), blk, 0, stream>>>(xg_buf, w2sf_whh, w2sf_bhh, doc_sen_lens,
                                                h_doc, 48, 0, 0);
  xg_gemm_kernel<<<dim3(3072, 3), blk, 0, stream>>>(w2sb_wih, w2sb_bih, emb, doc_tokens,
                                                    nullptr, doc_sen_lens, xg_buf, 48, 200, 1);
  gru_scan_kernel<<<dim3(64), blk, 0, stream>>>(xg_buf, w2sb_whh, w2sb_bhh, doc_sen_lens,
                                                h_doc, 48, 128, 1);
  attn_pool_kernel<<<dim3(1024), blk, 0, stream>>>(h_doc, doc_sen_lens, sen_proj_w, sen_proj_b,
                                                   sen_ctx, d_vec, 48);

  // ---- sentence-level BiGRU over d_vec (64 seqs, T=16, K=256)
  xg_gemm_kernel<<<dim3(64, 3), blk, 0, stream>>>(s2mf_wih, s2mf_bih, nullptr, nullptr,
                                                  d_vec, doc_lens, xg_buf, 16, 256, 0);
  gru_scan_kernel<<<dim3(4), blk, 0, stream>>>(xg_buf, s2mf_whh, s2mf_bhh, doc_lens,
                                               h_msg, 16, 0, 0);
  xg_gemm_kernel<<<dim3(64, 3), blk, 0, stream>>>(s2mb_wih, s2mb_bih, nullptr, nullptr,
                                                  d_vec, doc_lens, xg_buf, 16, 256, 1);
  gru_scan_kernel<<<dim3(4), blk, 0, stream>>>(xg_buf, s2mb_whh, s2mb_bhh, doc_lens,
                                               h_msg, 16, 128, 1);
  attn_pool_kernel<<<dim3(64), blk, 0, stream>>>(h_msg, doc_lens, msg_proj_w, msg_proj_b,
                                                 msg_ctx, msg_vec, 16);

  // ---- classifier head
  fc_hid_kernel<<<dim3(1), blk, 0, stream>>>(msg_vec, sent_vec, fc_w, fc_b, hid);
  logits_kernel<<<dim3(1), dim3(320), 0, stream>>>(hid, lik_w, lik_b, (float*)d_out);
}